// GCN_22222160789955
// MI455X (gfx1250) — compile-verified
//
#include <hip/hip_runtime.h>
#include <hip/hip_bf16.h>

typedef __attribute__((ext_vector_type(2))) float v2f;
typedef __attribute__((ext_vector_type(8))) float v8f;

// ---------------------------------------------------------------------------
// Degree / normalization kernels
// ---------------------------------------------------------------------------
__global__ void deg_init(float* __restrict__ deg, int n) {
    int i = blockIdx.x * blockDim.x + threadIdx.x;
    if (i < n) deg[i] = 1.0f;   // self-loop contributes 1
}

__global__ void deg_accum(const int* __restrict__ dst, float* __restrict__ deg, int e) {
    int i = blockIdx.x * blockDim.x + threadIdx.x;
    if (i < e) {
        __hip_atomic_fetch_add(&deg[dst[i]], 1.0f,
                               __ATOMIC_RELAXED, __HIP_MEMORY_SCOPE_AGENT);
    }
}

__global__ void deg_rsqrt(float* __restrict__ deg, int n) {
    int i = blockIdx.x * blockDim.x + threadIdx.x;
    if (i < n) deg[i] = rsqrtf(deg[i]);
}

// Zero-pad Wfc [64][10] -> Wpad [64][16] so the FC GEMM has branch-free loads
__global__ void pad_wfc(const float* __restrict__ Wfc, float* __restrict__ Wpad) {
    int i = blockIdx.x * blockDim.x + threadIdx.x;
    if (i < 64 * 16) {
        int k = i >> 4, n = i & 15;
        Wpad[i] = (n < 10) ? Wfc[k * 10 + n] : 0.0f;
    }
}

// ---------------------------------------------------------------------------
// fp32 WMMA GEMM:  C[M,NCOLS] = A[M,K] @ B[K,NCOLS]
// One wave computes a full 16-row x NCOLS strip: the A fragment for each
// K-step is loaded once and fed to NT = NCOLS/16 accumulators, so A traffic
// drops NT-fold and WMMA density per load rises.
// V_WMMA_F32_16X16X4_F32 operand layouts (ISA 7.12.2):
//   A 16x4:  lanes 0-15 hold K=k,k+1 ; lanes 16-31 hold K=k+2,k+3 (row = lane&15)
//   B 4x16:  vgpr0 = row K (lanes 0-15) / K+2 (lanes 16-31), col = lane&15
//   C 16x16: vgpr v -> row v (lanes 0-15) / v+8 (lanes 16-31), col = lane&15
// ---------------------------------------------------------------------------
template<int K, int NCOLS>
__global__ void gemm_wmma_f32(const float* __restrict__ A, const float* __restrict__ B,
                              float* __restrict__ C, int M) {
    constexpr int NT = NCOLS / 16;
    int wave = blockIdx.x * (blockDim.x >> 5) + (threadIdx.x >> 5);
    if (wave * 16 >= M) return;               // wave-uniform: EXEC stays all-ones

    int lane = threadIdx.x & 31;
    int half = lane >> 4;
    int r    = lane & 15;

    const float* Arow  = A + (size_t)(wave * 16 + r) * K;
    const float* Bbase = B + r;

    v8f acc[NT];
#pragma unroll
    for (int t = 0; t < NT; ++t) acc[t] = (v8f){};

#pragma unroll
    for (int k = 0; k < K; k += 4) {
        int kk = k + 2 * half;
        v2f a;
        a.x = Arow[kk];
        a.y = Arow[kk + 1];
        const float* B0 = Bbase + (size_t)kk * NCOLS;
#pragma unroll
        for (int t = 0; t < NT; ++t) {
            v2f b;
            b.x = B0[t * 16];
            b.y = B0[t * 16 + NCOLS];
            acc[t] = __builtin_amdgcn_wmma_f32_16x16x4_f32(false, a, false, b,
                                                           (short)0, acc[t], false, false);
        }
    }
#pragma unroll
    for (int t = 0; t < NT; ++t) {
#pragma unroll
        for (int v = 0; v < 8; ++v) {
            int m = wave * 16 + v + 8 * half;
            C[(size_t)m * NCOLS + t * 16 + r] = acc[t][v];
        }
    }
}

// FC head: out[M,10] = A[M,64] @ Wpad[64][16] + bias, storing only cols < 10
__global__ void gemm_fc_wmma(const float* __restrict__ A, const float* __restrict__ Wpad,
                             const float* __restrict__ bias, float* __restrict__ out,
                             int M) {
    int wave = blockIdx.x * (blockDim.x >> 5) + (threadIdx.x >> 5);
    if (wave * 16 >= M) return;

    int lane = threadIdx.x & 31;
    int half = lane >> 4;
    int r    = lane & 15;

    const float* Arow  = A + (size_t)(wave * 16 + r) * 64;
    const float* Bbase = Wpad + r;

    v8f c = {};
#pragma unroll
    for (int k = 0; k < 64; k += 4) {
        int kk = k + 2 * half;
        v2f a;
        a.x = Arow[kk];
        a.y = Arow[kk + 1];
        v2f b;
        b.x = Bbase[kk * 16];
        b.y = Bbase[kk * 16 + 16];
        c = __builtin_amdgcn_wmma_f32_16x16x4_f32(false, a, false, b,
                                                  (short)0, c, false, false);
    }
    if (r < 10) {                              // store mask only; all WMMAs done
        float bb = bias[r];
#pragma unroll
        for (int v = 0; v < 8; ++v) {
            int m = wave * 16 + v + 8 * half;
            out[(size_t)m * 10 + r] = c[v] + bb;
        }
    }
}

// ---------------------------------------------------------------------------
// Aggregation: agg starts with self-loop term, then atomic scatter over edges
// ---------------------------------------------------------------------------
template<int D>
__global__ void self_seed(const float* __restrict__ h, const float* __restrict__ dinv,
                          float* __restrict__ agg, int n) {
    size_t idx = (size_t)blockIdx.x * blockDim.x + threadIdx.x;
    if (idx < (size_t)n * D) {
        int node = (int)(idx / D);
        float di = dinv[node];
        agg[idx] = h[idx] * di * di;
    }
}

// One wave per edge, d=128: lane gathers float4 of h[src], 4 float atomics into agg[dst]
__global__ void scatter128(const int* __restrict__ src, const int* __restrict__ dst,
                           const float* __restrict__ dinv, const float* __restrict__ h,
                           float* __restrict__ agg, int E) {
    int e = blockIdx.x * (blockDim.x >> 5) + (threadIdx.x >> 5);
    if (e >= E) return;
    int lane = threadIdx.x & 31;
    int s = src[e];
    int t = dst[e];
    float cn = dinv[s] * dinv[t];
    float4 v = ((const float4*)(h + (size_t)s * 128))[lane];
    float* a = agg + (size_t)t * 128 + lane * 4;
    __hip_atomic_fetch_add(a + 0, v.x * cn, __ATOMIC_RELAXED, __HIP_MEMORY_SCOPE_AGENT);
    __hip_atomic_fetch_add(a + 1, v.y * cn, __ATOMIC_RELAXED, __HIP_MEMORY_SCOPE_AGENT);
    __hip_atomic_fetch_add(a + 2, v.z * cn, __ATOMIC_RELAXED, __HIP_MEMORY_SCOPE_AGENT);
    __hip_atomic_fetch_add(a + 3, v.w * cn, __ATOMIC_RELAXED, __HIP_MEMORY_SCOPE_AGENT);
}

// One wave per edge, d=64: lane gathers float2, 2 atomics
__global__ void scatter64(const int* __restrict__ src, const int* __restrict__ dst,
                          const float* __restrict__ dinv, const float* __restrict__ h,
                          float* __restrict__ agg, int E) {
    int e = blockIdx.x * (blockDim.x >> 5) + (threadIdx.x >> 5);
    if (e >= E) return;
    int lane = threadIdx.x & 31;
    int s = src[e];
    int t = dst[e];
    float cn = dinv[s] * dinv[t];
    float2 v = ((const float2*)(h + (size_t)s * 64))[lane];
    float* a = agg + (size_t)t * 64 + lane * 2;
    __hip_atomic_fetch_add(a + 0, v.x * cn, __ATOMIC_RELAXED, __HIP_MEMORY_SCOPE_AGENT);
    __hip_atomic_fetch_add(a + 1, v.y * cn, __ATOMIC_RELAXED, __HIP_MEMORY_SCOPE_AGENT);
}

template<int D>
__global__ void bias_relu(float* __restrict__ x, const float* __restrict__ b, int n) {
    size_t idx = (size_t)blockIdx.x * blockDim.x + threadIdx.x;
    if (idx < (size_t)n * D) {
        int f = (int)(idx % D);
        x[idx] = fmaxf(x[idx] + b[f], 0.0f);
    }
}

// ---------------------------------------------------------------------------
// Host-side launch
// ---------------------------------------------------------------------------
extern "C" void kernel_launch(void* const* d_in, const int* in_sizes, int n_in,
                              void* d_out, int out_size, void* d_ws, size_t ws_size,
                              hipStream_t stream) {
    const float* x   = (const float*)d_in[0];
    const int*   ei  = (const int*)d_in[1];
    const float* W1  = (const float*)d_in[2];
    const float* b1  = (const float*)d_in[3];
    const float* W2  = (const float*)d_in[4];
    const float* b2  = (const float*)d_in[5];
    const float* Wfc = (const float*)d_in[6];
    const float* bfc = (const float*)d_in[7];
    float* out = (float*)d_out;

    const int N = in_sizes[0] / 128;     // 100000
    const int E = in_sizes[1] / 2;       // 1600000
    const int* src = ei;
    const int* dst = ei + E;

    // Workspace: dinv[N pad] | bufA[N*128] | bufB[N*128] | Wpad[64*16]  (~103 MB)
    float* dinv = (float*)d_ws;
    float* bufA = dinv + (((size_t)N + 255) & ~(size_t)255);
    float* bufB = bufA + (size_t)N * 128;
    float* Wpad = bufB + (size_t)N * 128;

    const int TPB = 256;
    const int WPB = TPB / 32;   // 8 waves per block

    // --- degree & rsqrt normalization (+ pad Wfc for the FC head) ---
    deg_init<<<(N + TPB - 1) / TPB, TPB, 0, stream>>>(dinv, N);
    deg_accum<<<(E + TPB - 1) / TPB, TPB, 0, stream>>>(dst, dinv, E);
    deg_rsqrt<<<(N + TPB - 1) / TPB, TPB, 0, stream>>>(dinv, N);
    pad_wfc<<<(64 * 16 + TPB - 1) / TPB, TPB, 0, stream>>>(Wfc, Wpad);

    const int Mtiles = (N + 15) / 16;
    const int gemmBlocks = (Mtiles + WPB - 1) / WPB;   // one wave per 16-row strip

    // --- layer 1: h1 = x @ W1  (bufA), agg1 (bufB) ---
    {
        gemm_wmma_f32<128, 128><<<gemmBlocks, TPB, 0, stream>>>(x, W1, bufA, N);
        size_t tot = (size_t)N * 128;
        self_seed<128><<<(int)((tot + TPB - 1) / TPB), TPB, 0, stream>>>(bufA, dinv, bufB, N);
        scatter128<<<(E + WPB - 1) / WPB, TPB, 0, stream>>>(src, dst, dinv, bufA, bufB, E);
        bias_relu<128><<<(int)((tot + TPB - 1) / TPB), TPB, 0, stream>>>(bufB, b1, N);
    }

    // --- layer 2: h2 = relu(agg1) @ W2  (bufA[:N*64]), agg2 (bufB[:N*64]) ---
    {
        gemm_wmma_f32<128, 64><<<gemmBlocks, TPB, 0, stream>>>(bufB, W2, bufA, N);
        size_t tot = (size_t)N * 64;
        self_seed<64><<<(int)((tot + TPB - 1) / TPB), TPB, 0, stream>>>(bufA, dinv, bufB, N);
        scatter64<<<(E + WPB - 1) / WPB, TPB, 0, stream>>>(src, dst, dinv, bufA, bufB, E);
        bias_relu<64><<<(int)((tot + TPB - 1) / TPB), TPB, 0, stream>>>(bufB, b2, N);
    }

    // --- FC head: out = relu(agg2) @ Wpad + bfc ---
    gemm_fc_wmma<<<gemmBlocks, TPB, 0, stream>>>(bufB, Wpad, bfc, out, N);
}